// TimeSeriesLLM_83021717831749
// MI455X (gfx1250) — compile-verified
//
#include <hip/hip_runtime.h>
#include <hip/hip_bf16.h>
#include <math.h>

#define B_N 32
#define T_LEN 1024
#define C_IN 8
#define LH 32
#define K_NN 10
#define FUS_IN 96   // LH + TCN_OUT
#define TCN_OUT 64
#define FUS_H 256
#define REG_H 128
#define HOR 10

typedef __attribute__((ext_vector_type(2))) float v2f;
typedef __attribute__((ext_vector_type(8))) float v8f;

#if defined(__HIP_DEVICE_COMPILE__) && __has_builtin(__builtin_amdgcn_wmma_f32_16x16x4_f32)
#define USE_WMMA_F32 1
#else
#define USE_WMMA_F32 0
#endif

// ---------------------------------------------------------------------------
// h = x @ pw + pb  (per (b,t) row), plus sq = sum(h*h)
// ---------------------------------------------------------------------------
__global__ void k_embed(const float* __restrict__ x, const float* __restrict__ pw,
                        const float* __restrict__ pb, float* __restrict__ h,
                        float* __restrict__ sq) {
    int id = blockIdx.x * blockDim.x + threadIdx.x;
    if (id >= B_N * T_LEN) return;
    const float* xr = x + (size_t)id * C_IN;
    float xv[C_IN];
#pragma unroll
    for (int c = 0; c < C_IN; ++c) xv[c] = xr[c];
    float* hr = h + (size_t)id * LH;
    float s = 0.f;
#pragma unroll 4
    for (int j = 0; j < LH; ++j) {
        float a = pb[j];
#pragma unroll
        for (int c = 0; c < C_IN; ++c) a += xv[c] * pw[c * LH + j];
        hr[j] = a;
        s += a * a;
    }
    sq[id] = s;
}

// ---------------------------------------------------------------------------
// xT[b][c][t] = x[b][t][c]
// ---------------------------------------------------------------------------
__global__ void k_transpose(const float* __restrict__ x, float* __restrict__ xT) {
    int id = blockIdx.x * blockDim.x + threadIdx.x;
    if (id >= B_N * C_IN * T_LEN) return;
    int t = id % T_LEN;
    int c = (id / T_LEN) % C_IN;
    int b = id / (T_LEN * C_IN);
    xT[id] = x[((size_t)b * T_LEN + t) * C_IN + c];
}

// ---------------------------------------------------------------------------
// Laplace branch: pairwise d2 via fp32 WMMA, per-row top-10, smoothing,
// tanh, 32x32 matvec -> lapfeat (B,T,LH).  One wave (32 lanes) per 16 rows.
// ---------------------------------------------------------------------------
__global__ __launch_bounds__(32) void k_laplace(const float* __restrict__ h_all,
                                                const float* __restrict__ sq_all,
                                                const float* __restrict__ ow,
                                                const float* __restrict__ ob,
                                                float* __restrict__ lapfeat) {
    const int b    = blockIdx.y;
    const int t0   = blockIdx.x * 16;
    const int lane = threadIdx.x;

    const float* hB  = h_all + (size_t)b * T_LEN * LH;
    const float* sqB = sq_all + (size_t)b * T_LEN;

    __shared__ float lds_d2[16 * 16];
    __shared__ float lds_mv[32 * K_NN];
    __shared__ int   lds_mi[32 * K_NN];

    const int mrow = lane & 15;          // row (A) / col (B,D) index
    const int khi  = (lane >> 4) * 2;    // K sub-offset for upper half-wave
    const int moff = (lane >> 4) * 8;    // D-row offset for upper half-wave

    // A operand: h[t0+mrow][k0+khi .. +1] for 8 chained K-steps (K = LH = 32)
    v2f af[8];
#pragma unroll
    for (int s = 0; s < 8; ++s)
        af[s] = *(const v2f*)(hB + (size_t)(t0 + mrow) * LH + 4 * s + khi);

    float sqm[8];
#pragma unroll
    for (int v = 0; v < 8; ++v) sqm[v] = sqB[t0 + moff + v];

    float kv[K_NN];
    int   ki[K_NN];
#pragma unroll
    for (int j = 0; j < K_NN; ++j) { kv[j] = 3.0e38f; ki[j] = 0; }
    const int r_row  = lane >> 1;
    const int r_half = (lane & 1) * 8;

    for (int s0 = 0; s0 < T_LEN; s0 += 16) {
        const float sqn = sqB[s0 + mrow];
        v8f acc = {};
#if USE_WMMA_F32
#pragma unroll
        for (int s = 0; s < 8; ++s) {
            v2f bf = *(const v2f*)(hB + (size_t)(s0 + mrow) * LH + 4 * s + khi);
            acc = __builtin_amdgcn_wmma_f32_16x16x4_f32(
                false, af[s], false, bf, (short)0, acc, false, false);
        }
#pragma unroll
        for (int v = 0; v < 8; ++v) {
            int m = moff + v;
            float d2 = sqm[v] + sqn - 2.0f * acc[v];
            d2 = fmaxf(d2, 0.0f);
            if (t0 + m == s0 + mrow) d2 = 1.0e18f;
            lds_d2[m * 16 + mrow] = d2;
        }
#else
#pragma unroll
        for (int v = 0; v < 8; ++v) {
            int m = moff + v;
            const float* hr = hB + (size_t)(t0 + m) * LH;
            const float* hc = hB + (size_t)(s0 + mrow) * LH;
            float dot = 0.f;
#pragma unroll
            for (int j = 0; j < LH; ++j) dot += hr[j] * hc[j];
            float d2 = sqm[v] + sqn - 2.0f * dot;
            d2 = fmaxf(d2, 0.0f);
            if (t0 + m == s0 + mrow) d2 = 1.0e18f;
            lds_d2[m * 16 + mrow] = d2;
        }
        (void)acc;
#endif
        __syncthreads();

        // top-K insertion (stable: strict <, keeps earliest index on ties)
#pragma unroll
        for (int j = 0; j < 8; ++j) {
            float d  = lds_d2[r_row * 16 + r_half + j];
            int   si = s0 + r_half + j;
            if (d < kv[K_NN - 1]) {
                kv[K_NN - 1] = d; ki[K_NN - 1] = si;
#pragma unroll
                for (int q = K_NN - 1; q > 0; --q) {
                    if (kv[q] < kv[q - 1]) {
                        float tv = kv[q]; kv[q] = kv[q - 1]; kv[q - 1] = tv;
                        int   ti = ki[q]; ki[q] = ki[q - 1]; ki[q - 1] = ti;
                    }
                }
            }
        }
        __syncthreads();
    }

#pragma unroll
    for (int j = 0; j < K_NN; ++j) {
        lds_mv[lane * K_NN + j] = kv[j];
        lds_mi[lane * K_NN + j] = ki[j];
    }
    __syncthreads();

    if (lane < 16) {
        const float* va = &lds_mv[(lane * 2 + 0) * K_NN];
        const int*   ia = &lds_mi[(lane * 2 + 0) * K_NN];
        const float* vb = &lds_mv[(lane * 2 + 1) * K_NN];
        const int*   ib = &lds_mi[(lane * 2 + 1) * K_NN];
        int mi[K_NN];
        int pa = 0, pb2 = 0;
#pragma unroll
        for (int j = 0; j < K_NN; ++j) {
            float A = va[pa], Bv = vb[pb2];
            bool take_a = (A < Bv) || (A == Bv && ia[pa] < ib[pb2]);
            if (take_a) { mi[j] = ia[pa]; ++pa; }
            else        { mi[j] = ib[pb2]; ++pb2; }
        }

        const int t = t0 + lane;
        float hr[LH], sm[LH];
#pragma unroll
        for (int j = 0; j < LH; ++j) { hr[j] = hB[(size_t)t * LH + j]; sm[j] = 0.f; }
        float wsum = 0.f;
        const float inv2sig = 1.0f / (2.0f * 1.0f * 1.0f + 1e-8f);
#pragma unroll
        for (int kq = 0; kq < K_NN; ++kq) {
            const float* hn = hB + (size_t)mi[kq] * LH;
            float dd = 0.f;
#pragma unroll
            for (int j = 0; j < LH; ++j) { float df = hn[j] - hr[j]; dd += df * df; }
            float w = expf(-dd * inv2sig);
            wsum += w;
#pragma unroll
            for (int j = 0; j < LH; ++j) sm[j] += w * hn[j];
        }
        float inv = 1.0f / (wsum + 1e-8f);
        float tn[LH];
#pragma unroll
        for (int j = 0; j < LH; ++j) tn[j] = tanhf(hr[j] - sm[j] * inv);

        float* outp = lapfeat + ((size_t)b * T_LEN + t) * LH;
#pragma unroll 4
        for (int o = 0; o < LH; ++o) {
            float a = ob[o];
#pragma unroll
            for (int j = 0; j < LH; ++j) a += tn[j] * ow[j * LH + o];
            outp[o] = a;
        }
    }
}

// ---------------------------------------------------------------------------
// fused[b][j] = mean_t lapfeat[b][t][j]   (j < LH, cols 0..31 of fused)
// ---------------------------------------------------------------------------
__global__ void k_pool(const float* __restrict__ lapfeat, float* __restrict__ fused) {
    int b = blockIdx.x, j = blockIdx.y;
    __shared__ float red[256];
    float s = 0.f;
    for (int t = threadIdx.x; t < T_LEN; t += 256)
        s += lapfeat[((size_t)b * T_LEN + t) * LH + j];
    red[threadIdx.x] = s;
    __syncthreads();
    for (int st = 128; st > 0; st >>= 1) {
        if (threadIdx.x < st) red[threadIdx.x] += red[threadIdx.x + st];
        __syncthreads();
    }
    if (threadIdx.x == 0) fused[b * FUS_IN + j] = red[0] * (1.0f / T_LEN);
}

// ---------------------------------------------------------------------------
// Causal dilated conv as implicit-im2col GEMM on fp32 WMMA.
// Templated on (CIN, KS, DIL) so the K-chain fully unrolls; the im2col tile
// B[n][k] is staged once in LDS per (b, t-tile) block and reused across all
// Cout tiles, so the inner loop is pure: global_b64(w) + ds_b64(B) + wmma.
// ---------------------------------------------------------------------------
template <int CIN, int KKC, int DIL>
__global__ __launch_bounds__(32) void k_conv_t(const float* __restrict__ xin_all,
                                               const float* __restrict__ w,
                                               const float* __restrict__ bias,
                                               const float* __restrict__ addend_all,
                                               float* __restrict__ out_all,
                                               int Cout, int doRelu) {
    constexpr int KFLAT = CIN * KKC;     // always a multiple of 4
    constexpr int KROW  = KFLAT + 2;     // pad (even) to spread LDS banks
    const int b    = blockIdx.y;
    const int t0   = blockIdx.x * 16;
    const int lane = threadIdx.x;
    const int mrow = lane & 15;
    const int khi  = (lane >> 4) * 2;
    const int moff = (lane >> 4) * 8;
    const float* xin = xin_all + (size_t)b * CIN * T_LEN;

    __shared__ float ldsB[16 * KROW];

    // cooperative im2col fill: ldsB[n][k] = xin[ci, t0+n-(KKC-1-kr)*DIL] (0 pad)
    for (int idx = lane; idx < 16 * KFLAT; idx += 32) {
        int n = idx / KFLAT;
        int k = idx - n * KFLAT;
        int ci, tt;
        if (KKC == 3) { ci = k / 3; int kr = k - 3 * ci; tt = t0 + n - (2 - kr) * DIL; }
        else          { ci = k;     tt = t0 + n; }
        ldsB[n * KROW + k] = (tt >= 0) ? xin[(size_t)ci * T_LEN + tt] : 0.0f;
    }
    __syncthreads();

    float* yout = out_all + (size_t)b * Cout * T_LEN;
    const float* addp = addend_all ? addend_all + (size_t)b * Cout * T_LEN : nullptr;

    for (int co0 = 0; co0 < Cout; co0 += 16) {
        v8f acc = {};
#if USE_WMMA_F32
#pragma unroll
        for (int s = 0; s < KFLAT / 4; ++s) {
            v2f a  = *(const v2f*)(w + (size_t)(co0 + mrow) * KFLAT + 4 * s + khi);
            v2f bf = *(const v2f*)(&ldsB[mrow * KROW + 4 * s + khi]);
            acc = __builtin_amdgcn_wmma_f32_16x16x4_f32(
                false, a, false, bf, (short)0, acc, false, false);
        }
#else
#pragma unroll
        for (int v = 0; v < 8; ++v) {
            int m = moff + v;
            float sacc = 0.f;
            for (int k = 0; k < KFLAT; ++k)
                sacc += w[(size_t)(co0 + m) * KFLAT + k] * ldsB[mrow * KROW + k];
            acc[v] = sacc;
        }
#endif
#pragma unroll
        for (int v = 0; v < 8; ++v) {
            int m  = co0 + moff + v;
            int tn = t0 + mrow;
            float val = acc[v] + bias[m];
            if (addp) val += addp[(size_t)m * T_LEN + tn];
            if (doRelu) val = fmaxf(val, 0.0f);
            yout[(size_t)m * T_LEN + tn] = val;
        }
    }
}

// ---------------------------------------------------------------------------
// Dense GEMM on WMMA: out(B x OUT) = relu?(in(B x IN) @ w(IN x OUT) + bias)
// grid = (OUT/16, B/16); A rows = batch rows (contiguous float2 loads).
// ---------------------------------------------------------------------------
template <int IN>
__global__ __launch_bounds__(32) void k_dense_wmma(const float* __restrict__ in,
                                                   const float* __restrict__ w,
                                                   const float* __restrict__ bias,
                                                   float* __restrict__ out,
                                                   int OUT, int doRelu) {
    const int o0   = blockIdx.x * 16;
    const int b0   = blockIdx.y * 16;
    const int lane = threadIdx.x;
    const int mrow = lane & 15;
    const int khi  = (lane >> 4) * 2;
    const int moff = (lane >> 4) * 8;

    v8f acc = {};
#if USE_WMMA_F32
#pragma unroll
    for (int s = 0; s < IN / 4; ++s) {
        v2f a = *(const v2f*)(in + (size_t)(b0 + mrow) * IN + 4 * s + khi);
        v2f bf;
        bf[0] = w[(size_t)(4 * s + khi + 0) * OUT + o0 + mrow];
        bf[1] = w[(size_t)(4 * s + khi + 1) * OUT + o0 + mrow];
        acc = __builtin_amdgcn_wmma_f32_16x16x4_f32(
            false, a, false, bf, (short)0, acc, false, false);
    }
#else
#pragma unroll
    for (int v = 0; v < 8; ++v) {
        int m = b0 + moff + v;
        float sacc = 0.f;
        for (int k = 0; k < IN; ++k)
            sacc += in[(size_t)m * IN + k] * w[(size_t)k * OUT + o0 + mrow];
        acc[v] = sacc;
    }
#endif
#pragma unroll
    for (int v = 0; v < 8; ++v) {
        int m = b0 + moff + v;     // batch row
        int o = o0 + mrow;
        float val = acc[v] + bias[o];
        if (doRelu) val = fmaxf(val, 0.0f);
        out[(size_t)m * OUT + o] = val;
    }
}

// ---------------------------------------------------------------------------
// tcn head: fused[b][32+o] = head_b[o] + sum_ci y2[b,ci,T-1]*head_w[ci,o]
// ---------------------------------------------------------------------------
__global__ void k_head(const float* __restrict__ y2, const float* __restrict__ hw,
                       const float* __restrict__ hb, float* __restrict__ fused) {
    int b = blockIdx.x, o = threadIdx.x;  // 64 threads
    float a = hb[o];
    const float* yb = y2 + (size_t)b * 128 * T_LEN;
    for (int ci = 0; ci < 128; ++ci)
        a += yb[(size_t)ci * T_LEN + (T_LEN - 1)] * hw[ci * TCN_OUT + o];
    fused[b * FUS_IN + LH + o] = a;
}

// ---------------------------------------------------------------------------
// scalar dense (only for OUT=10 final head)
// ---------------------------------------------------------------------------
__global__ void k_dense(const float* __restrict__ in, const float* __restrict__ w,
                        const float* __restrict__ bias, float* __restrict__ out,
                        int IN, int OUT, int doRelu) {
    int b = blockIdx.x;
    extern __shared__ float sx[];
    for (int i = threadIdx.x; i < IN; i += blockDim.x) sx[i] = in[(size_t)b * IN + i];
    __syncthreads();
    for (int o = threadIdx.x; o < OUT; o += blockDim.x) {
        float a = bias[o];
        for (int i = 0; i < IN; ++i) a += sx[i] * w[(size_t)i * OUT + o];
        out[(size_t)b * OUT + o] = doRelu ? fmaxf(a, 0.0f) : a;
    }
}

// ---------------------------------------------------------------------------
extern "C" void kernel_launch(void* const* d_in, const int* in_sizes, int n_in,
                              void* d_out, int out_size, void* d_ws, size_t ws_size,
                              hipStream_t stream) {
    const float* x  = (const float*)d_in[0];
    const float* pw = (const float*)d_in[1];
    const float* pb = (const float*)d_in[2];
    const float* ow = (const float*)d_in[3];
    const float* ob = (const float*)d_in[4];
    const float* bw[3][6];
    for (int i = 0; i < 3; ++i)
        for (int j = 0; j < 6; ++j) bw[i][j] = (const float*)d_in[5 + i * 6 + j];
    const float* head_w = (const float*)d_in[23];
    const float* head_b = (const float*)d_in[24];
    const float* f1w = (const float*)d_in[25];
    const float* f1b = (const float*)d_in[26];
    const float* f2w = (const float*)d_in[27];
    const float* f2b = (const float*)d_in[28];
    const float* r1w = (const float*)d_in[29];
    const float* r1b = (const float*)d_in[30];
    const float* r2w = (const float*)d_in[31];
    const float* r2b = (const float*)d_in[32];

    float* ws = (float*)d_ws;
    size_t off = 0;
    float* h       = ws + off; off += (size_t)B_N * T_LEN * LH;
    float* sq      = ws + off; off += (size_t)B_N * T_LEN;
    float* lapfeat = ws + off; off += (size_t)B_N * T_LEN * LH;
    float* xT      = ws + off; off += (size_t)B_N * C_IN * T_LEN;
    float* bufA    = ws + off; off += (size_t)B_N * 128 * T_LEN;
    float* bufB    = ws + off; off += (size_t)B_N * 128 * T_LEN;
    float* y0      = ws + off; off += (size_t)B_N * 32 * T_LEN;
    float* y1      = ws + off; off += (size_t)B_N * 64 * T_LEN;
    float* y2      = ws + off; off += (size_t)B_N * 128 * T_LEN;
    float* fused   = ws + off; off += (size_t)B_N * FUS_IN;
    float* f1o     = ws + off; off += (size_t)B_N * FUS_H;
    float* f2o     = ws + off; off += (size_t)B_N * FUS_H;
    float* r1o     = ws + off; off += (size_t)B_N * REG_H;

    k_embed<<<(B_N * T_LEN + 255) / 256, 256, 0, stream>>>(x, pw, pb, h, sq);
    k_transpose<<<(B_N * C_IN * T_LEN + 255) / 256, 256, 0, stream>>>(x, xT);

    k_laplace<<<dim3(T_LEN / 16, B_N), 32, 0, stream>>>(h, sq, ow, ob, lapfeat);
    k_pool<<<dim3(B_N, LH), 256, 0, stream>>>(lapfeat, fused);

    const dim3 cgrid(T_LEN / 16, B_N);
    // block 0: 8 -> 32, dil 1
    k_conv_t<8, 3, 1><<<cgrid, 32, 0, stream>>>(xT,   bw[0][0], bw[0][1], nullptr, bufA, 32, 1);
    k_conv_t<32, 3, 1><<<cgrid, 32, 0, stream>>>(bufA, bw[0][2], bw[0][3], nullptr, bufB, 32, 0);
    k_conv_t<8, 1, 1><<<cgrid, 32, 0, stream>>>(xT,   bw[0][4], bw[0][5], bufB,    y0,   32, 1);
    // block 1: 32 -> 64, dil 2
    k_conv_t<32, 3, 2><<<cgrid, 32, 0, stream>>>(y0,   bw[1][0], bw[1][1], nullptr, bufA, 64, 1);
    k_conv_t<64, 3, 2><<<cgrid, 32, 0, stream>>>(bufA, bw[1][2], bw[1][3], nullptr, bufB, 64, 0);
    k_conv_t<32, 1, 1><<<cgrid, 32, 0, stream>>>(y0,   bw[1][4], bw[1][5], bufB,    y1,   64, 1);
    // block 2: 64 -> 128, dil 4
    k_conv_t<64, 3, 4><<<cgrid, 32, 0, stream>>>(y1,   bw[2][0], bw[2][1], nullptr, bufA, 128, 1);
    k_conv_t<128, 3, 4><<<cgrid, 32, 0, stream>>>(bufA, bw[2][2], bw[2][3], nullptr, bufB, 128, 0);
    k_conv_t<64, 1, 1><<<cgrid, 32, 0, stream>>>(y1,   bw[2][4], bw[2][5], bufB,    y2,   128, 1);

    k_head<<<B_N, TCN_OUT, 0, stream>>>(y2, head_w, head_b, fused);

    k_dense_wmma<FUS_IN><<<dim3(FUS_H / 16, B_N / 16), 32, 0, stream>>>(fused, f1w, f1b, f1o, FUS_H, 1);
    k_dense_wmma<FUS_H><<<dim3(FUS_H / 16, B_N / 16), 32, 0, stream>>>(f1o, f2w, f2b, f2o, FUS_H, 1);
    k_dense_wmma<FUS_H><<<dim3(REG_H / 16, B_N / 16), 32, 0, stream>>>(f2o, r1w, r1b, r1o, REG_H, 1);
    k_dense<<<B_N, 256, REG_H * sizeof(float), stream>>>(r1o, r2w, r2b, (float*)d_out, REG_H, HOR, 0);
}